// DisenVAE_4063039062652
// MI455X (gfx1250) — compile-verified
//
#include <hip/hip_runtime.h>
#include <hip/hip_bf16.h>
#include <math.h>

// ---------------- problem constants ----------------
constexpr int MM   = 20000;   // items
constexpr int KK   = 7;       // cores
constexpr int DD   = 100;     // latent dim
constexpr int HH   = 300;     // hidden
constexpr int NN   = 1024;    // batch
constexpr float TAUF = 0.1f;
constexpr int NR   = NN * KK; // 7168 rows of Xk / T / h

// workspace layout (floats)
constexpr long WS_CORESN = 0;                         // 7*100 -> pad 704
constexpr long WS_ITEMSN = 704;                       // 20000*100
constexpr long WS_CATES  = WS_ITEMSN + (long)MM * DD; // 20000*7
constexpr long WS_T      = WS_CATES + (long)MM * KK;  // 7168*300
constexpr long WS_ZT     = WS_T + (long)NR * HH;      // 7168*100

// output layout (floats)
constexpr long OUT_LOGITS = 0;
constexpr long OUT_MU     = (long)NN * MM;            // 20,480,000
constexpr long OUT_LV     = OUT_MU + (long)NR * DD;   // +716,800

// LDS strides
constexpr int XS_STRIDE = 34;   // A-tile row stride: even (b64-aligned pairs), conflict-free
constexpr int BP_STRIDE = 160;  // B pair-row stride (floats): 32*p bank offset between rows

// ---------------- WMMA helper (fp32 16x16x4) ----------------
typedef float v2f __attribute__((ext_vector_type(2)));
typedef float v8f __attribute__((ext_vector_type(8)));

__device__ __forceinline__ v8f wmma_f32(v2f a, v2f b, v8f c) {
  // (neg_a, A, neg_b, B, c_mod, C, reuse_a, reuse_b)
  return __builtin_amdgcn_wmma_f32_16x16x4_f32(false, a, false, b, (short)0, c,
                                               false, false);
}

__device__ __forceinline__ v2f lds_pair(const float* p) {
  return *(const v2f*)p;  // 8-byte aligned by construction
}

__device__ __forceinline__ float wave_sum(float s) {
#pragma unroll
  for (int off = 16; off; off >>= 1) s += __shfl_xor(s, off, 32);
  return s;
}

// ---------------- kernel A: cores_n ----------------
__global__ void k_cores(const float* __restrict__ cores, float* __restrict__ cores_n) {
  int wave = threadIdx.x >> 5, lane = threadIdx.x & 31;
  if (wave >= KK) return;
  float v[4]; float s = 0.f;
#pragma unroll
  for (int j = 0; j < 4; ++j) {
    int d = lane + 32 * j;
    v[j] = (d < DD) ? cores[wave * DD + d] : 0.f;
    s += v[j] * v[j];
  }
  s = wave_sum(s);
  float inv = 1.f / fmaxf(sqrtf(s), 1e-12f);
#pragma unroll
  for (int j = 0; j < 4; ++j) {
    int d = lane + 32 * j;
    if (d < DD) cores_n[wave * DD + d] = v[j] * inv;
  }
}

// ---------------- kernel B: items_n + cates (softmax over K) ----------------
__global__ void k_items(const float* __restrict__ items,
                        const float* __restrict__ cores_n,
                        float* __restrict__ items_n,
                        float* __restrict__ cates) {
  int wave = threadIdx.x >> 5, lane = threadIdx.x & 31;
  int m = blockIdx.x * 8 + wave;
  if (m >= MM) return;
  float v[4]; float s = 0.f;
#pragma unroll
  for (int j = 0; j < 4; ++j) {
    int d = lane + 32 * j;
    v[j] = (d < DD) ? items[(long)m * DD + d] : 0.f;
    s += v[j] * v[j];
  }
  s = wave_sum(s);
  float inv = 1.f / fmaxf(sqrtf(s), 1e-12f);
#pragma unroll
  for (int j = 0; j < 4; ++j) {
    int d = lane + 32 * j;
    v[j] *= inv;
    if (d < DD) items_n[(long)m * DD + d] = v[j];
  }
  float dot[KK];
#pragma unroll
  for (int k = 0; k < KK; ++k) {
    float p = 0.f;
#pragma unroll
    for (int j = 0; j < 4; ++j) {
      int d = lane + 32 * j;
      if (d < DD) p += v[j] * cores_n[k * DD + d];
    }
    dot[k] = wave_sum(p) / TAUF;
  }
  float mx = dot[0];
#pragma unroll
  for (int k = 1; k < KK; ++k) mx = fmaxf(mx, dot[k]);
  float e[KK]; float ssum = 0.f;
#pragma unroll
  for (int k = 0; k < KK; ++k) { e[k] = expf(dot[k] - mx); ssum += e[k]; }
  if (lane == 0) {
#pragma unroll
    for (int k = 0; k < KK; ++k) cates[(long)m * KK + k] = e[k] / ssum;
  }
}

// ---------------- kernel C: T = tanh(X @ (cates_k * W1) + b1) ----------------
// grid: (hblocks=5, iblocks=16, k=7), block 256
__global__ void k_gemm1(const float* __restrict__ X, const float* __restrict__ W1,
                        const float* __restrict__ b1, const float* __restrict__ cates,
                        float* __restrict__ T) {
  const int k  = blockIdx.z;
  const int i0 = blockIdx.y * 64;
  const int h0 = blockIdx.x * 64;
  __shared__ __align__(16) float Xs[64 * XS_STRIDE];   // 64 rows x 32 k
  __shared__ __align__(16) float Bp[16 * BP_STRIDE];   // 16 pair-rows x 64 cols (float2)
  __shared__ float cs[32];

  const int t = threadIdx.x, wave = t >> 5, lane = t & 31;
  const int rw = wave & 3, cw = wave >> 2;          // 4x2 wave grid
  const int rowBase = rw * 16, colBase = cw * 32;
  const int hs = lane >> 4, l = lane & 15;
  const int aRow = (rowBase + l) * XS_STRIDE + 2 * hs;

  v8f acc0 = {}; v8f acc1 = {};

  for (int m0 = 0; m0 < MM; m0 += 32) {
    // prefetch next chunk while this one is being staged/consumed
    if (m0 + 32 < MM) {
      if (t < 64) {
        __builtin_prefetch(&X[(long)(i0 + t) * MM + m0 + 32], 0, 1);
      } else if (t < 128) {
        int j = t - 64, r = j >> 1, part = j & 1;
        int h = h0 + part * 32;
        if (h < HH) __builtin_prefetch(&W1[(long)(m0 + 32 + r) * HH + h], 0, 1);
      }
    }
    __syncthreads();
    if (t < 32) cs[t] = cates[(long)(m0 + t) * KK + k];
#pragma unroll 2
    for (int idx = t; idx < 64 * 32; idx += 256) {
      int r = idx >> 5, c = idx & 31;
      Xs[r * XS_STRIDE + c] = X[(long)(i0 + r) * MM + m0 + c];
    }
    __syncthreads();
#pragma unroll 2
    for (int idx = t; idx < 32 * 64; idx += 256) {
      int r = idx >> 6, c = idx & 63;   // r = k-row, c = col
      int h = h0 + c;
      float w = (h < HH) ? W1[(long)(m0 + r) * HH + h] : 0.f;
      Bp[(r >> 1) * BP_STRIDE + 2 * c + (r & 1)] = w * cs[r];
    }
    __syncthreads();
#pragma unroll
    for (int kk = 0; kk < 32; kk += 4) {
      int p = (kk >> 1) + hs;           // pair-row for this half-wave
      v2f a  = lds_pair(&Xs[aRow + kk]);
      v2f bA = lds_pair(&Bp[p * BP_STRIDE + 2 * (colBase + l)]);
      acc0 = wmma_f32(a, bA, acc0);
      v2f bB = lds_pair(&Bp[p * BP_STRIDE + 2 * (colBase + 16 + l)]);
      acc1 = wmma_f32(a, bB, acc1);
    }
  }
  // epilogue: bias + tanh, store into T[(i*7+k)*300 + h]
  int hA = h0 + colBase + l;
  int hB = hA + 16;
  float bA = (hA < HH) ? b1[hA] : 0.f;
  float bB = (hB < HH) ? b1[hB] : 0.f;
#pragma unroll
  for (int v = 0; v < 8; ++v) {
    int irow = i0 + rowBase + v + 8 * hs;
    long rT = ((long)irow * KK + k) * HH;
    if (hA < HH) T[rT + hA] = tanhf(acc0[v] + bA);
    if (hB < HH) T[rT + hB] = tanhf(acc1[v] + bB);
  }
}

// ---------------- kernel D: h = T @ W2 + b2 -> mu/logvar ----------------
// grid 448 (16 rows each), block 256 (threads 0..199 active in compute)
__global__ void k_gemm2(const float* __restrict__ T, const float* __restrict__ W2,
                        const float* __restrict__ b2, float* __restrict__ out) {
  const int r0 = blockIdx.x * 16;
  __shared__ float Ts[16 * 300];
  const int t = threadIdx.x;
  for (int idx = t; idx < 16 * 300; idx += 256)
    Ts[idx] = T[(long)r0 * HH + idx];
  __syncthreads();
  if (t < 2 * DD) {
    float acc[16];
#pragma unroll
    for (int rr = 0; rr < 16; ++rr) acc[rr] = 0.f;
    for (int d3 = 0; d3 < HH; ++d3) {
      float w = W2[(long)d3 * (2 * DD) + t];
#pragma unroll
      for (int rr = 0; rr < 16; ++rr) acc[rr] += Ts[rr * HH + d3] * w;
    }
    float bb = b2[t];
#pragma unroll
    for (int rr = 0; rr < 16; ++rr) {
      int r = r0 + rr;
      float val = acc[rr] + bb;
      if (t < DD) out[OUT_MU + (long)r * DD + t] = val;
      else        out[OUT_LV + (long)r * DD + (t - DD)] = val;
    }
  }
}

// ---------------- kernel E: zt = l2norm(mu) / TAU ----------------
__global__ void k_zt(const float* __restrict__ out, float* __restrict__ zt) {
  int wave = threadIdx.x >> 5, lane = threadIdx.x & 31;
  int r = blockIdx.x * 8 + wave;
  if (r >= NR) return;
  const float* mu = out + OUT_MU + (long)r * DD;
  float v[4]; float s = 0.f;
#pragma unroll
  for (int j = 0; j < 4; ++j) {
    int d = lane + 32 * j;
    v[j] = (d < DD) ? mu[d] : 0.f;
    s += v[j] * v[j];
  }
  s = wave_sum(s);
  float inv = 1.f / (fmaxf(sqrtf(s), 1e-12f) * TAUF);
#pragma unroll
  for (int j = 0; j < 4; ++j) {
    int d = lane + 32 * j;
    if (d < DD) zt[(long)r * DD + d] = v[j] * inv;
  }
}

// ---------------- kernel F: probs = sum_k exp(zt @ items_n^T) * cates ----------------
// grid: (mblocks=313, iblocks=64), block 256. Block tile: 112 rows (16 i x 7 k) x 64 m.
constexpr int AS_STRIDE = 102;  // even -> b64-aligned A pairs; rows distinct mod 64
__global__ void k_probs(const float* __restrict__ zt, const float* __restrict__ items_n,
                        const float* __restrict__ cates, float* __restrict__ probs) {
  const int i0 = blockIdx.y * 16;
  const int m0 = blockIdx.x * 64;
  const int r0 = i0 * KK;  // 112 consecutive rows
  __shared__ __align__(16) float As[112 * AS_STRIDE];  // 45.7 KB (zt tile; later exp tile)
  __shared__ __align__(16) float Bp[16 * BP_STRIDE];   // 10.2 KB (items_n chunk, pair rows)

  const int t = threadIdx.x, wave = t >> 5, lane = t & 31;
  const int hs = lane >> 4, l = lane & 31 & 15;
  const int mvalid = min(64, MM - m0);

  for (int idx = t; idx < 112 * DD; idx += 256) {
    int r = idx / DD, d = idx % DD;
    As[r * AS_STRIDE + d] = zt[(long)(r0 + r) * DD + d];
  }

  v8f acc[4] = {{}, {}, {}, {}};  // up to 4 of the 28 (7 rowtiles x 4 coltiles) tiles

  for (int kc0 = 0; kc0 < DD; kc0 += 32) {
    int kw = min(32, DD - kc0);  // 32,32,32,4
    __syncthreads();
    for (int idx = t; idx < 64 * 32; idx += 256) {
      int c = idx >> 5, dd = idx & 31;
      if (dd < kw)
        Bp[(dd >> 1) * BP_STRIDE + 2 * c + (dd & 1)] =
            (c < mvalid) ? items_n[(long)(m0 + c) * DD + kc0 + dd] : 0.f;
    }
    __syncthreads();
#pragma unroll
    for (int ti = 0; ti < 4; ++ti) {
      int tile = wave + 8 * ti;
      if (tile < 28) {
        int rt = tile >> 2, ct = tile & 3;
        for (int kk = 0; kk < kw; kk += 4) {
          int p = (kk >> 1) + hs;
          v2f a = lds_pair(&As[(rt * 16 + l) * AS_STRIDE + kc0 + kk + 2 * hs]);
          v2f b = lds_pair(&Bp[p * BP_STRIDE + 2 * (ct * 16 + l)]);
          acc[ti] = wmma_f32(a, b, acc[ti]);
        }
      }
    }
  }
  __syncthreads();
  float* Es = As;  // overlay: 112 x 64 exp tile
#pragma unroll
  for (int ti = 0; ti < 4; ++ti) {
    int tile = wave + 8 * ti;
    if (tile < 28) {
      int rt = tile >> 2, ct = tile & 3;
#pragma unroll
      for (int v = 0; v < 8; ++v) {
        int lr = rt * 16 + v + 8 * hs;
        int c = ct * 16 + l;
        Es[lr * 64 + c] = expf(acc[ti][v]);
      }
    }
  }
  __syncthreads();
  // weighted 7-way reduction -> probs (deterministic, no atomics)
  for (int idx = t; idx < 16 * 64; idx += 256) {
    int ii = idx >> 6, c = idx & 63;
    if (c < mvalid) {
      float p = 0.f;
#pragma unroll
      for (int k = 0; k < KK; ++k)
        p += Es[(ii * KK + k) * 64 + c] * cates[(long)(m0 + c) * KK + k];
      probs[(long)(i0 + ii) * MM + m0 + c] = p;
    }
  }
}

// ---------------- kernel G: logits = log(p) - log(sum p) in place ----------------
__global__ void k_logsoftmax(float* __restrict__ logits) {
  __shared__ float red[256];
  __shared__ float lsum;
  const int t = threadIdx.x;
  long base = (long)blockIdx.x * MM;
  float s = 0.f;
  for (int m = t; m < MM; m += 256) s += logits[base + m];
  red[t] = s;
  __syncthreads();
#pragma unroll
  for (int off = 128; off; off >>= 1) {
    if (t < off) red[t] += red[t + off];
    __syncthreads();
  }
  if (t == 0) lsum = logf(red[0]);
  __syncthreads();
  float ls = lsum;
  for (int m = t; m < MM; m += 256) {
    float p = logits[base + m];
    logits[base + m] = logf(p) - ls;
  }
}

// ---------------- launcher ----------------
extern "C" void kernel_launch(void* const* d_in, const int* in_sizes, int n_in,
                              void* d_out, int out_size, void* d_ws, size_t ws_size,
                              hipStream_t stream) {
  const float* X     = (const float*)d_in[0];
  // d_in[1] = A (int32, unused by reference math)
  const float* W1    = (const float*)d_in[2];
  const float* b1    = (const float*)d_in[3];
  const float* W2    = (const float*)d_in[4];
  const float* b2    = (const float*)d_in[5];
  const float* items = (const float*)d_in[6];
  const float* cores = (const float*)d_in[7];
  float* out = (float*)d_out;
  float* ws  = (float*)d_ws;

  float* cores_n = ws + WS_CORESN;
  float* items_n = ws + WS_ITEMSN;
  float* cates   = ws + WS_CATES;
  float* T       = ws + WS_T;
  float* zt      = ws + WS_ZT;

  k_cores<<<1, 256, 0, stream>>>(cores, cores_n);
  k_items<<<(MM + 7) / 8, 256, 0, stream>>>(items, cores_n, items_n, cates);
  k_gemm1<<<dim3((HH + 63) / 64, NN / 64, KK), 256, 0, stream>>>(X, W1, b1, cates, T);
  k_gemm2<<<NR / 16, 256, 0, stream>>>(T, W2, b2, out);
  k_zt<<<(NR + 7) / 8, 256, 0, stream>>>(out, zt);
  k_probs<<<dim3((MM + 63) / 64, NN / 16), 256, 0, stream>>>(zt, items_n, cates, out);
  k_logsoftmax<<<NN, 256, 0, stream>>>(out);
}